// SphericalHarmonicsFunction_35820027249310
// MI455X (gfx1250) — compile-verified
//
#include <hip/hip_runtime.h>
#include <hip/hip_bf16.h>

typedef float v2f __attribute__((ext_vector_type(2)));
typedef float v4f __attribute__((ext_vector_type(4)));
typedef float v8f __attribute__((ext_vector_type(8)));

// Z[m][k]: harmonic m (sympy Znm order, max_l=4) as linear combo of 16 monomials
// t = [1, x, x2, x3, s*sp, s*cp, x*s*sp, x*s*cp, x2*s*sp, x2*s*cp,
//      s2*s2p, s2*c2p, x*s2*s2p, x*s2*c2p, s3*s3p, s3*c3p]
__device__ __constant__ float ZM[256] = {
 0.28209479177387814f,0.f,0.f,0.f, 0.f,0.f,0.f,0.f, 0.f,0.f,0.f,0.f, 0.f,0.f,0.f,0.f,
 0.f,0.f,0.f,0.f, -0.48860251190291992f,0.f,0.f,0.f, 0.f,0.f,0.f,0.f, 0.f,0.f,0.f,0.f,
 0.f,0.48860251190291992f,0.f,0.f, 0.f,0.f,0.f,0.f, 0.f,0.f,0.f,0.f, 0.f,0.f,0.f,0.f,
 0.f,0.f,0.f,0.f, 0.f,-0.48860251190291992f,0.f,0.f, 0.f,0.f,0.f,0.f, 0.f,0.f,0.f,0.f,
 0.f,0.f,0.f,0.f, 0.f,0.f,0.f,0.f, 0.f,0.f,-0.54627421529603959f,0.f, 0.f,0.f,0.f,0.f,
 0.f,0.f,0.f,0.f, 0.f,0.f,-1.09254843059207918f,0.f, 0.f,0.f,0.f,0.f, 0.f,0.f,0.f,0.f,
 -0.31539156525252005f,0.f,0.94617469575756008f,0.f, 0.f,0.f,0.f,0.f, 0.f,0.f,0.f,0.f, 0.f,0.f,0.f,0.f,
 0.f,0.f,0.f,0.f, 0.f,0.f,0.f,-1.09254843059207918f, 0.f,0.f,0.f,0.f, 0.f,0.f,0.f,0.f,
 0.f,0.f,0.f,0.f, 0.f,0.f,0.f,0.f, 0.f,0.f,0.f,0.54627421529603959f, 0.f,0.f,0.f,0.f,
 0.f,0.f,0.f,0.f, 0.f,0.f,0.f,0.f, 0.f,0.f,0.f,0.f, 0.f,0.f,-0.59004358992664352f,0.f,
 0.f,0.f,0.f,0.f, 0.f,0.f,0.f,0.f, 0.f,0.f,0.f,0.f, -1.44530572132027735f,0.f,0.f,0.f,
 0.f,0.f,0.f,0.f, 0.45704579946446577f,0.f,0.f,0.f, -2.28522899732232886f,0.f,0.f,0.f, 0.f,0.f,0.f,0.f,
 0.f,-1.11952899777035953f,0.f,1.86588166295059921f, 0.f,0.f,0.f,0.f, 0.f,0.f,0.f,0.f, 0.f,0.f,0.f,0.f,
 0.f,0.f,0.f,0.f, 0.f,0.45704579946446577f,0.f,0.f, 0.f,-2.28522899732232886f,0.f,0.f, 0.f,0.f,0.f,0.f,
 0.f,0.f,0.f,0.f, 0.f,0.f,0.f,0.f, 0.f,0.f,0.f,0.f, 0.f,1.44530572132027735f,0.f,0.f,
 0.f,0.f,0.f,0.f, 0.f,0.f,0.f,0.f, 0.f,0.f,0.f,0.f, 0.f,0.f,0.f,-0.59004358992664352f
};

#define TPW 4  // tiles (of 16 angles) per wave in the bulk kernel

// One wave32 handles TILES full tiles of 16 angles. Per tile:
// D(16harm x 16ang) = Z(16x16) * T(16mono x 16ang) via 4 chained
// v_wmma_f32_16x16x4_f32. D layout maps onto row-major [angle][16]:
// lane L<16 -> harmonics 0..7 of angle L, lane L>=16 -> harmonics 8..15.
// Launch guarantees every surviving wave owns TILES full tiles:
// no interior guards, EXEC stays all-1s (WMMA requirement), and all per-tile
// addresses are immediate offsets from base pointers.
template <int TILES>
__global__ __launch_bounds__(256) void sh_znm_wmma_kernel(
    const float* __restrict__ cos_theta, const float* __restrict__ phi,
    float* __restrict__ out, int numWaves, int tileBase, int wavesPerBlk) {
  const int lane = threadIdx.x & 31;
  // wave id as a scalar -> top guard is s_cmp/s_cbranch, no EXEC manipulation
  const int waveInBlk = __builtin_amdgcn_readfirstlane((int)threadIdx.x >> 5);
  const int waveId = (int)blockIdx.x * wavesPerBlk + waveInBlk;
  if (waveId >= numWaves) return;

  const bool hf = (lane & 16) != 0;  // upper half-wave: K={2,3} / M={8..15}
  const int l16 = lane & 15;
  const long tile0 = (long)tileBase + (long)waveId * TILES;
  const long angle0 = tile0 * 16 + l16;

  const float* ctp = cos_theta + angle0;
  const float* php = phi + angle0;
  float* outp = out + angle0 * 16 + (hf ? 8 : 0);

  // A operands (constant Z, 16x4 chunks), resident across all tiles.
  // ksel even -> 8B-aligned v2f loads (global_load_b64).
  const int arow = l16 * 16 + (hf ? 2 : 0);
  const v2f a0 = *reinterpret_cast<const v2f*>(&ZM[arow + 0]);
  const v2f a1 = *reinterpret_cast<const v2f*>(&ZM[arow + 4]);
  const v2f a2 = *reinterpret_cast<const v2f*>(&ZM[arow + 8]);
  const v2f a3 = *reinterpret_cast<const v2f*>(&ZM[arow + 12]);

#pragma unroll
  for (int i = 0; i < TILES; ++i) {
    const float x = ctp[i * 16];
    const float p = php[i * 16];
    // raw v_sqrt_f32 (~1 ULP): arg in [0,1], no denorm/class handling needed
    const float s = __builtin_amdgcn_sqrtf(__builtin_fmaxf(1.0f - x * x, 0.0f));
    float sp, cp;
    __sincosf(p, &sp, &cp);
    const float s2p = 2.0f * sp * cp;
    const float c2p = 1.0f - 2.0f * sp * sp;
    const float s3p = sp * c2p + cp * s2p;
    const float c3p = cp * c2p - sp * s2p;
    const float x2 = x * x, x3 = x2 * x;
    const float ss = s * s, sss = ss * s;

    const float ssp   = s * sp,    scp   = s * cp;
    const float xssp  = x * ssp,   xscp  = x * scp;
    const float x2ssp = x * xssp,  x2scp = x * xscp;
    const float ss2p  = ss * s2p,  sc2p  = ss * c2p;
    const float xss2p = x * ss2p,  xsc2p = x * sc2p;
    const float s3s3p = sss * s3p, s3c3p = sss * c3p;

    // B operands: lower half K={4j,4j+1}, upper half K={4j+2,4j+3}
    v2f b0, b1, b2, b3;
    b0.x = hf ? x2    : 1.0f;   b0.y = hf ? x3    : x;
    b1.x = hf ? xssp  : ssp;    b1.y = hf ? xscp  : scp;
    b2.x = hf ? ss2p  : x2ssp;  b2.y = hf ? sc2p  : x2scp;
    b3.x = hf ? s3s3p : xss2p;  b3.y = hf ? s3c3p : xsc2p;

    v8f acc = {};
    acc = __builtin_amdgcn_wmma_f32_16x16x4_f32(false, a0, false, b0, (short)0, acc, false, false);
    acc = __builtin_amdgcn_wmma_f32_16x16x4_f32(false, a1, false, b1, (short)0, acc, false, false);
    acc = __builtin_amdgcn_wmma_f32_16x16x4_f32(false, a2, false, b2, (short)0, acc, false, false);
    acc = __builtin_amdgcn_wmma_f32_16x16x4_f32(false, a3, false, b3, (short)0, acc, false, false);

    v4f lo = {acc[0], acc[1], acc[2], acc[3]};
    v4f hi = {acc[4], acc[5], acc[6], acc[7]};
    __builtin_nontemporal_store(lo, (v4f*)(outp + i * 256));
    __builtin_nontemporal_store(hi, (v4f*)(outp + i * 256 + 4));
  }
}

// Scalar tail for the (n % 16) leftover angles; not launched when n is a
// multiple of 16 (the harness case: n = 2^23).
__global__ void sh_znm_tail_kernel(const float* __restrict__ cos_theta,
                                   const float* __restrict__ phi,
                                   float* __restrict__ out, int n, int start) {
  const int i = start + blockIdx.x * blockDim.x + threadIdx.x;
  if (i >= n) return;
  const float x = cos_theta[i];
  const float p = phi[i];
  const float s = __builtin_amdgcn_sqrtf(__builtin_fmaxf(1.0f - x * x, 0.0f));
  float sp, cp;
  __sincosf(p, &sp, &cp);
  const float s2p = 2.0f * sp * cp;
  const float c2p = 1.0f - 2.0f * sp * sp;
  const float s3p = sp * c2p + cp * s2p;
  const float c3p = cp * c2p - sp * s2p;
  const float x2 = x * x, x3 = x2 * x;
  const float ss = s * s, sss = ss * s;
  const float ssp = s * sp, scp = s * cp;

  float* o = out + (long)i * 16;
  o[0]  = 0.28209479177387814f;
  o[1]  = -0.48860251190291992f * ssp;
  o[2]  = 0.48860251190291992f * x;
  o[3]  = -0.48860251190291992f * scp;
  o[4]  = -0.54627421529603959f * ss * s2p;
  o[5]  = -1.09254843059207918f * x * ssp;
  o[6]  = 0.94617469575756008f * x2 - 0.31539156525252005f;
  o[7]  = -1.09254843059207918f * x * scp;
  o[8]  = 0.54627421529603959f * ss * c2p;
  o[9]  = -0.59004358992664352f * sss * s3p;
  o[10] = -1.44530572132027735f * x * ss * s2p;
  o[11] = (0.45704579946446577f - 2.28522899732232886f * x2) * ssp;
  o[12] = 1.86588166295059921f * x3 - 1.11952899777035953f * x;
  o[13] = (0.45704579946446577f - 2.28522899732232886f * x2) * scp;
  o[14] = 1.44530572132027735f * x * ss * c2p;
  o[15] = -0.59004358992664352f * sss * c3p;
}

extern "C" void kernel_launch(void* const* d_in, const int* in_sizes, int n_in,
                              void* d_out, int out_size, void* d_ws, size_t ws_size,
                              hipStream_t stream) {
  const float* cos_theta = (const float*)d_in[0];
  const float* phi       = (const float*)d_in[1];
  float* out             = (float*)d_out;
  const int n = in_sizes[0];

  const int fullTiles = n / 16;        // 16-angle tiles that use the WMMA path
  const int bulkWaves = fullTiles / TPW;
  if (bulkWaves > 0) {                 // each wave: TPW full tiles, no guards
    const int wavesPerBlock = 256 / 32;
    const int blocks = (bulkWaves + wavesPerBlock - 1) / wavesPerBlock;
    sh_znm_wmma_kernel<TPW><<<blocks, 256, 0, stream>>>(
        cos_theta, phi, out, bulkWaves, 0, wavesPerBlock);
  }
  const int remTiles = fullTiles - bulkWaves * TPW;   // 0..3 leftover tiles
  if (remTiles > 0) {                  // one wave per leftover tile
    sh_znm_wmma_kernel<1><<<1, remTiles * 32, 0, stream>>>(
        cos_theta, phi, out, remTiles, bulkWaves * TPW, remTiles);
  }
  const int rem = n - fullTiles * 16;  // < 16 leftover angles
  if (rem > 0) {
    sh_znm_tail_kernel<<<1, 32, 0, stream>>>(cos_theta, phi, out, n, fullTiles * 16);
  }
}